// _GNN_46145128628993
// MI455X (gfx1250) — compile-verified
//
#include <hip/hip_runtime.h>
#include <hip/hip_bf16.h>

#define N_NODES 100000
#define N_EDGES 800000
#define D_MODEL 128

typedef __attribute__((ext_vector_type(2))) float v2f;
typedef __attribute__((ext_vector_type(8))) float v8f;

// ---------------------------------------------------------------------------
// Zero-fill (float4 vectorized).  n must be a multiple of 4.
// ---------------------------------------------------------------------------
__global__ __launch_bounds__(256) void k_zero(float* __restrict__ p, int n4) {
    int i = blockIdx.x * blockDim.x + threadIdx.x;
    if (i < n4) ((float4*)p)[i] = make_float4(0.f, 0.f, 0.f, 0.f);
}

// ---------------------------------------------------------------------------
// Weighted degree: deg[r[i]] += e[i]; deg[s[i]] += e[i]   (undirected graph)
// ---------------------------------------------------------------------------
__global__ __launch_bounds__(256) void k_degree(const int* __restrict__ S,
                                                const int* __restrict__ R,
                                                const float* __restrict__ E,
                                                float* __restrict__ deg, int nE) {
    int i = blockIdx.x * blockDim.x + threadIdx.x;
    if (i >= nE) return;
    float e = E[i];
    unsafeAtomicAdd(&deg[R[i]], e);
    unsafeAtomicAdd(&deg[S[i]], e);
}

// ---------------------------------------------------------------------------
// dis[i] = deg>0 ? rsqrt(deg) : 0
// ---------------------------------------------------------------------------
__global__ __launch_bounds__(256) void k_dis(const float* __restrict__ deg,
                                             float* __restrict__ dis, int n) {
    int i = blockIdx.x * blockDim.x + threadIdx.x;
    if (i >= n) return;
    float d = deg[i];
    dis[i] = (d > 0.f) ? rsqrtf(d) : 0.f;
}

// ---------------------------------------------------------------------------
// Y[M x 128] = X[M x 128] @ W[128 x 128] using V_WMMA_F32_16X16X4_F32.
// Block = 256 threads (8 wave32s); blockIdx.x = M-tile, wave id = N-tile.
// A layout (16x4 f32): lanes 0-15 = rows M0..15 with K = {0,1} in v0/v1,
//                      lanes 16-31 = same rows with K = {2,3}.
// B layout (4x16 f32): mirrored with N across lanes.
// C/D layout: VGPR v, lanes 0-15 -> (M=v, N=lane); lanes 16-31 -> (M=8+v).
// ---------------------------------------------------------------------------
__global__ __launch_bounds__(256) void k_gemm_wmma(const float* __restrict__ X,
                                                   const float* __restrict__ W,
                                                   float* __restrict__ Y) {
    const int lane  = threadIdx.x & 31;
    const int wave  = threadIdx.x >> 5;          // N-tile 0..7
    const int mtile = blockIdx.x;                // 6250 tiles of 16 rows
    const int l15   = lane & 15;
    const int khalf = lane >> 4;                 // 0 or 1

    const int row = mtile * 16 + l15;            // A row for this lane
    const int col = wave * 16 + l15;             // B/D column for this lane

    const float* __restrict__ xrow = X + (size_t)row * D_MODEL;

    v8f acc = {};
#pragma unroll
    for (int k = 0; k < D_MODEL; k += 4) {
        const int ka = k + 2 * khalf;            // this lane's K pair
        v2f a, b;
        a.x = xrow[ka];
        a.y = xrow[ka + 1];
        b.x = W[(size_t)ka * D_MODEL + col];
        b.y = W[(size_t)(ka + 1) * D_MODEL + col];
        acc = __builtin_amdgcn_wmma_f32_16x16x4_f32(
            /*neg_a=*/false, a, /*neg_b=*/false, b,
            /*c_mod=*/(short)0, acc, /*reuse_a=*/false, /*reuse_b=*/false);
    }

    const int mbase = mtile * 16 + (khalf ? 8 : 0);
#pragma unroll
    for (int v = 0; v < 8; ++v)
        Y[(size_t)(mbase + v) * D_MODEL + col] = acc[v];
}

// ---------------------------------------------------------------------------
// Edge scatter: one wave per directed edge.  w = dis[s]*e*dis[r].
// Lane l handles features 4l..4l+3: float4 gather from xW[s], 4 fp32
// global atomic adds into agg[r] (hardware global_atomic_add_f32).
// Directed edges 0..nE-1 use (S,R); nE..2nE-1 use (R,S).
// ---------------------------------------------------------------------------
__global__ __launch_bounds__(256) void k_scatter(const float* __restrict__ xW,
                                                 const int* __restrict__ S,
                                                 const int* __restrict__ R,
                                                 const float* __restrict__ E,
                                                 const float* __restrict__ dis,
                                                 float* __restrict__ agg, int nE) {
    const int gwave = (blockIdx.x * blockDim.x + threadIdx.x) >> 5;
    const int lane  = threadIdx.x & 31;
    if (gwave >= 2 * nE) return;

    const bool fwd = gwave < nE;
    const int  i   = fwd ? gwave : gwave - nE;
    const int  s   = fwd ? S[i] : R[i];
    const int  r   = fwd ? R[i] : S[i];
    const float w  = dis[s] * E[i] * dis[r];

    const float4 m = ((const float4*)(xW + (size_t)s * D_MODEL))[lane];
    float* dst = agg + (size_t)r * D_MODEL + lane * 4;
    unsafeAtomicAdd(dst + 0, m.x * w);
    unsafeAtomicAdd(dst + 1, m.y * w);
    unsafeAtomicAdd(dst + 2, m.z * w);
    unsafeAtomicAdd(dst + 3, m.w * w);
}

// ---------------------------------------------------------------------------
// out = x + agg + b   (float4 vectorized; 128 floats/row => 32 float4/row)
// Safe in-place when out == x (pure elementwise).
// ---------------------------------------------------------------------------
__global__ __launch_bounds__(256) void k_combine(const float* __restrict__ x,
                                                 const float* __restrict__ agg,
                                                 const float* __restrict__ b,
                                                 float* __restrict__ out, int n4) {
    int i = blockIdx.x * blockDim.x + threadIdx.x;
    if (i >= n4) return;
    const int bcol4 = i & (D_MODEL / 4 - 1);     // float4 column within row
    float4 xv = ((const float4*)x)[i];
    float4 av = ((const float4*)agg)[i];
    float4 bv = ((const float4*)b)[bcol4];
    float4 o;
    o.x = xv.x + av.x + bv.x;
    o.y = xv.y + av.y + bv.y;
    o.z = xv.z + av.z + bv.z;
    o.w = xv.w + av.w + bv.w;
    ((float4*)out)[i] = o;
}

// ---------------------------------------------------------------------------
extern "C" void kernel_launch(void* const* d_in, const int* in_sizes, int n_in,
                              void* d_out, int out_size, void* d_ws, size_t ws_size,
                              hipStream_t stream) {
    const float* nodes = (const float*)d_in[0];
    const int*   S     = (const int*)  d_in[1];
    const int*   R     = (const int*)  d_in[2];
    const float* E     = (const float*)d_in[3];
    const float* W1    = (const float*)d_in[4];
    const float* b1    = (const float*)d_in[5];
    const float* W2    = (const float*)d_in[6];
    const float* b2    = (const float*)d_in[7];
    float* out = (float*)d_out;

    // Workspace layout (floats): xW | agg | deg | dis
    float* xW  = (float*)d_ws;
    float* agg = xW  + (size_t)N_NODES * D_MODEL;
    float* deg = agg + (size_t)N_NODES * D_MODEL;
    float* dis = deg + N_NODES;

    const int nd   = N_NODES * D_MODEL;          // 12.8M floats
    const int nd4  = nd / 4;
    const int degB = (N_NODES + 1023) / 1024;    // zero deg in float4s
    const int mt   = N_NODES / 16;               // 6250 M-tiles (exact)

    // --- shared normalization (same for both layers) ---
    k_zero  <<<degB, 256, 0, stream>>>(deg, N_NODES / 4);
    k_degree<<<(N_EDGES + 255) / 256, 256, 0, stream>>>(S, R, E, deg, N_EDGES);
    k_dis   <<<(N_NODES + 255) / 256, 256, 0, stream>>>(deg, dis, N_NODES);

    const int scatterBlocks = (2 * N_EDGES * 32 + 255) / 256;   // wave per edge

    // --- layer 1:  h1 = nodes + gcn(nodes, W1, b1)  -> stored in d_out ---
    k_gemm_wmma<<<mt, 256, 0, stream>>>(nodes, W1, xW);
    k_zero     <<<(nd4 + 255) / 256, 256, 0, stream>>>(agg, nd4);
    k_scatter  <<<scatterBlocks, 256, 0, stream>>>(xW, S, R, E, dis, agg, N_EDGES);
    k_combine  <<<(nd4 + 255) / 256, 256, 0, stream>>>(nodes, agg, b1, out, nd4);

    // --- layer 2:  out = h1 + gcn(h1, W2, b2)  (in place on d_out) ---
    k_gemm_wmma<<<mt, 256, 0, stream>>>(out, W2, xW);
    k_zero     <<<(nd4 + 255) / 256, 256, 0, stream>>>(agg, nd4);
    k_scatter  <<<scatterBlocks, 256, 0, stream>>>(xW, S, R, E, dis, agg, N_EDGES);
    k_combine  <<<(nd4 + 255) / 256, 256, 0, stream>>>(out, agg, b2, out, nd4);
}